// KeyAct_12472585028076
// MI455X (gfx1250) — compile-verified
//
#include <hip/hip_runtime.h>
#include <cstdint>

// ---------------------------------------------------------------------------
// KeyAct pair loss for MI455X (gfx1250, wave32)
//   - response tensor (96 MB) is L2-resident (192 MB L2)
//   - one workgroup per pair; both 128 KB patches DMA'd into LDS, each with a
//     SINGLE CDNA5 TDM tensor_load_to_lds (3D descriptor: 16 x 16 x 128 tile,
//     strides W=256 and T*H*W=196608), tracked by TENSORcnt.
//   - pass 1: block min/max of each patch (wave32 shfl_xor + LDS tree)
//   - pass 2: sum of |n1-n2|^pw via v_log_f32 / v_exp_f32
//   - second kernel reduces the 2048 per-pair values deterministically.
// ---------------------------------------------------------------------------

namespace {
constexpr int kC = 128, kT = 3, kH = 256, kW = 256, kP = 16;
constexpr int kPatch = kC * kP * kP;   // 32768 floats = 128 KB
constexpr int kVec4  = kPatch / 4;     // 8192 float4s
constexpr int kBlk   = 256;            // 8 wave32 waves
constexpr int kWaves = kBlk / 32;
constexpr int kCS    = kT * kH * kW;   // channel stride (floats)
}

typedef unsigned int u32x4 __attribute__((ext_vector_type(4)));
typedef unsigned int u32x8 __attribute__((ext_vector_type(8)));

// One TDM descriptor-driven DMA: 3D tile [z=128][y=16][x=16] of f32 from
// global (strides: x contiguous, y stride kW, z stride kCS) into LDS,
// packed linearly as c*256 + py*16 + px.
__device__ __forceinline__ void tdm_load_patch(unsigned lds_off, const float* gbase) {
  const unsigned long long ga = (unsigned long long)(uintptr_t)gbase;

  u32x4 g0;
  g0[0] = 1u;                                   // count=1, no gather, user D#
  g0[1] = lds_off;                              // lds_addr (bytes)
  g0[2] = (unsigned)(ga & 0xFFFFFFFFu);         // global_addr[31:0]
  g0[3] = (unsigned)((ga >> 32) & 0x01FFFFFFu)  // global_addr[56:32]
          | 0x80000000u;                        // type=2 ("image")

  // tensor dims set large (tile always in-bounds); strides exact.
  const unsigned td0 = 0x7FFFFFFFu, td1 = 0x7FFFFFFFu;
  const unsigned long long s0 = (unsigned long long)kW * 4ull / 4ull;   // dim0 stride (elems) = 256
  const unsigned long long s1 = (unsigned long long)kCS;                // dim1 stride (elems) = 196608
  u32x8 g1;
  g1[0] = 0x00020000u;                          // wg_mask=0, data_size=2 (4 B)
  g1[1] = (td0 & 0xFFFFu) << 16;                // [47:32]=0, tensor_dim0[15:0]
  g1[2] = (td0 >> 16) | ((td1 & 0xFFFFu) << 16);// tensor_dim0[31:16], tensor_dim1[15:0]
  g1[3] = (td1 >> 16) | (16u << 16);            // tensor_dim1[31:16], tile_dim0=16
  g1[4] = 16u | (128u << 16);                   // tile_dim1=16, tile_dim2=128
  g1[5] = (unsigned)(s0 & 0xFFFFFFFFull);       // tensor_dim0_stride[31:0]
  g1[6] = (unsigned)(s0 >> 32)                  // tensor_dim0_stride[47:32]
          | ((unsigned)(s1 & 0xFFFFull) << 16); // tensor_dim1_stride[15:0]
  g1[7] = (unsigned)(s1 >> 16);                 // tensor_dim1_stride[47:16]

  u32x4 g2;
  g2[0] = 0x7FFFFFFFu;                          // tensor_dim2 (large)
  g2[1] = 0u;                                   // tensor_dim3
  g2[2] = 0u;                                   // tensor_dim2_stride[31:0]
  g2[3] = 0u;                                   // stride hi, tile_dim3=0 (unused)

  u32x4 g3 = {0u, 0u, 0u, 0u};                  // dim3 stride / dim4 / tile_dim4 = 0

  asm volatile("tensor_load_to_lds %0, %1, %2, %3"
               :: "s"(g0), "s"(g1), "s"(g2), "s"(g3)
               : "memory");
}

__device__ __forceinline__ void wait_tensor_zero() {
#if __has_builtin(__builtin_amdgcn_s_wait_tensorcnt)
  __builtin_amdgcn_s_wait_tensorcnt(0);
#else
  asm volatile("s_wait_tensorcnt 0x0" ::: "memory");
#endif
}

__global__ __launch_bounds__(kBlk) void pair_loss_kernel(
    const float* __restrict__ resp,
    const int* __restrict__ ys1, const int* __restrict__ xs1, const int* __restrict__ ts1,
    const int* __restrict__ ys2, const int* __restrict__ xs2, const int* __restrict__ ts2,
    const int* __restrict__ ptype, const float* __restrict__ powers,
    const float* __restrict__ plus, float* __restrict__ per_out)
{
  __shared__ float patch1[kPatch];          // 128 KB
  __shared__ float patch2[kPatch];          // 128 KB
  __shared__ float rbuf[4][kWaves];
  __shared__ float ssum[kWaves];
  __shared__ float bcast[6];                // mn1, inv1, norm1?, mn2, inv2, norm2?

  const int pair = blockIdx.x;
  const int tid  = threadIdx.x;

  const int t1 = ts1[pair], y1 = ys1[pair], x1 = xs1[pair];
  const int t2 = ts2[pair], y2 = ys2[pair], x2 = xs2[pair];
  const float* base1 = resp + (size_t)t1 * (kH * kW) + (size_t)y1 * kW + (size_t)x1;
  const float* base2 = resp + (size_t)t2 * (kH * kW) + (size_t)y2 * kW + (size_t)x2;

  // Low 32 bits of the flat shared-aperture address are the LDS byte offset.
  const unsigned lds1 = (unsigned)(uintptr_t)&patch1[0];
  const unsigned lds2 = (unsigned)(uintptr_t)&patch2[0];

  // ---- TDM: one descriptor DMA per 128 KB patch; wave 0 issues and waits ----
  if (tid == 0) {
    tdm_load_patch(lds1, base1);
    tdm_load_patch(lds2, base2);
    wait_tensor_zero();
  }
  __syncthreads();

  const float4* p1 = reinterpret_cast<const float4*>(patch1);
  const float4* p2 = reinterpret_cast<const float4*>(patch2);

  // ---- pass 1: min/max of each patch ----
  float mn1 = 3.4e38f, mx1 = -3.4e38f, mn2 = 3.4e38f, mx2 = -3.4e38f;
  for (int v = tid; v < kVec4; v += kBlk) {
    float4 a = p1[v], b = p2[v];
    mn1 = fminf(mn1, fminf(fminf(a.x, a.y), fminf(a.z, a.w)));
    mx1 = fmaxf(mx1, fmaxf(fmaxf(a.x, a.y), fmaxf(a.z, a.w)));
    mn2 = fminf(mn2, fminf(fminf(b.x, b.y), fminf(b.z, b.w)));
    mx2 = fmaxf(mx2, fmaxf(fmaxf(b.x, b.y), fmaxf(b.z, b.w)));
  }
  #pragma unroll
  for (int off = 16; off > 0; off >>= 1) {
    mn1 = fminf(mn1, __shfl_xor(mn1, off, 32));
    mx1 = fmaxf(mx1, __shfl_xor(mx1, off, 32));
    mn2 = fminf(mn2, __shfl_xor(mn2, off, 32));
    mx2 = fmaxf(mx2, __shfl_xor(mx2, off, 32));
  }
  const int wave = tid >> 5, lane = tid & 31;
  if (lane == 0) {
    rbuf[0][wave] = mn1; rbuf[1][wave] = mx1;
    rbuf[2][wave] = mn2; rbuf[3][wave] = mx2;
  }
  __syncthreads();
  if (tid == 0) {
    float a0 = rbuf[0][0], a1 = rbuf[1][0], a2 = rbuf[2][0], a3 = rbuf[3][0];
    for (int i = 1; i < kWaves; ++i) {
      a0 = fminf(a0, rbuf[0][i]); a1 = fmaxf(a1, rbuf[1][i]);
      a2 = fminf(a2, rbuf[2][i]); a3 = fmaxf(a3, rbuf[3][i]);
    }
    const float r1 = a1 - a0, r2 = a3 - a2;
    bcast[0] = a0; bcast[1] = (r1 != 0.f) ? 1.f / r1 : 0.f; bcast[2] = (r1 != 0.f) ? 1.f : 0.f;
    bcast[3] = a2; bcast[4] = (r2 != 0.f) ? 1.f / r2 : 0.f; bcast[5] = (r2 != 0.f) ? 1.f : 0.f;
  }
  __syncthreads();

  const float mnA = bcast[0], invA = bcast[1]; const bool nrmA = bcast[2] != 0.f;
  const float mnB = bcast[3], invB = bcast[4]; const bool nrmB = bcast[5] != 0.f;

  const int   pt = ptype[pair];
  const float pw = powers[pt];

  // ---- pass 2: sum |n1 - n2|^pw  (exp2(pw*log2 t); t==0 -> exp2(-inf)=0) ----
  float acc = 0.f;
  for (int v = tid; v < kVec4; v += kBlk) {
    float4 a = p1[v], b = p2[v];
    float av[4] = {a.x, a.y, a.z, a.w};
    float bv[4] = {b.x, b.y, b.z, b.w};
    #pragma unroll
    for (int j = 0; j < 4; ++j) {
      const float na = nrmA ? (av[j] - mnA) * invA : av[j];
      const float nb = nrmB ? (bv[j] - mnB) * invB : bv[j];
      const float t  = fabsf(na - nb);
      // |t|^pw = exp2(pw * log2 t) on the HW transcendental unit
      acc += __builtin_amdgcn_exp2f(pw * __builtin_amdgcn_logf(t));
    }
  }
  #pragma unroll
  for (int off = 16; off > 0; off >>= 1)
    acc += __shfl_xor(acc, off, 32);
  if (lane == 0) ssum[wave] = acc;
  __syncthreads();
  if (tid == 0) {
    float s = ssum[0];
    for (int i = 1; i < kWaves; ++i) s += ssum[i];
    const float d = s * (1.f / (float)kPatch);
    float per = (pt == 2) ? d : 1.f / (1.f + d);
    if (pt != 0) per += 0.5f * plus[0];
    per_out[pair] = per;
  }
}

__global__ __launch_bounds__(kBlk) void reduce_mean_kernel(
    const float* __restrict__ per, int n, float* __restrict__ out)
{
  __shared__ float s[kBlk];
  float acc = 0.f;
  for (int i = threadIdx.x; i < n; i += kBlk) acc += per[i];
  s[threadIdx.x] = acc;
  __syncthreads();
  for (int off = kBlk / 2; off > 0; off >>= 1) {
    if ((int)threadIdx.x < off) s[threadIdx.x] += s[threadIdx.x + off];
    __syncthreads();
  }
  if (threadIdx.x == 0) out[0] = s[0] / (float)n;
}

extern "C" void kernel_launch(void* const* d_in, const int* in_sizes, int n_in,
                              void* d_out, int out_size, void* d_ws, size_t ws_size,
                              hipStream_t stream) {
  const float* resp   = (const float*)d_in[0];
  const int*   ys1    = (const int*)d_in[1];
  const int*   xs1    = (const int*)d_in[2];
  const int*   ts1    = (const int*)d_in[3];
  const int*   ys2    = (const int*)d_in[4];
  const int*   xs2    = (const int*)d_in[5];
  const int*   ts2    = (const int*)d_in[6];
  const int*   ptype  = (const int*)d_in[7];
  const float* powers = (const float*)d_in[8];
  const float* plus   = (const float*)d_in[9];
  const int N = in_sizes[1];

  float* per = (float*)d_ws;   // N floats of scratch

  pair_loss_kernel<<<N, kBlk, 0, stream>>>(resp, ys1, xs1, ts1, ys2, xs2, ts2,
                                           ptype, powers, plus, per);
  reduce_mean_kernel<<<1, kBlk, 0, stream>>>(per, N, (float*)d_out);
}